// Encoder_68607807586564
// MI455X (gfx1250) — compile-verified
//
#include <hip/hip_runtime.h>
#include <hip/hip_bf16.h>

typedef float v2f __attribute__((ext_vector_type(2)));
typedef float v8f __attribute__((ext_vector_type(8)));

#define FG_KCHUNK 256     // K per block in final gemm

// ---------------------------------------------------------------------------
// Spiral conv + ELU:  y[b,n,co] = elu( sum_k gather(x)[b,n,k] * W[k,co] + bias[co] )
// gather(x)[b,n, s*C+c] = x[b, idx[n,s], c],  K = S*C (zero-padded to K4)
// One block = one 16-row tile (rows share b, consecutive n; Nv is 2^nvShift).
// Waves split the Cout/16 column tiles; each runs V_WMMA_F32_16X16X4_F32 over K
// with two independent accumulator chains for pipelining.
// ---------------------------------------------------------------------------
__global__ void spiral_conv_wmma(const float* __restrict__ x,    // [B, Nv, C]
                                 const int*   __restrict__ idx,  // [Nv, S]
                                 const float* __restrict__ W,    // [K, Cout]
                                 const float* __restrict__ bias, // [Cout]
                                 float* __restrict__ y,          // [B, Nv, Cout]
                                 int nvShift, int C, int S, int Cout, int K4)
{
    extern __shared__ float Alds[];            // [16][K4]
    const int Nv = 1 << nvShift;
    const int K = S * C;
    const int rowBase = blockIdx.x << 4;       // global row in [0, B*Nv)
    const int b  = rowBase >> nvShift;         // whole 16-row tile shares b
    const int n0 = rowBase & (Nv - 1);

    // cooperative gather into LDS (zero-pad K..K4)
    for (int e = threadIdx.x; e < 16 * K4; e += blockDim.x) {
        const int r = e / K4, k = e % K4;
        float v = 0.0f;
        if (k < K) {
            const int s = k / C, c = k % C;
            const int src = idx[(n0 + r) * S + s];
            v = x[((size_t)b << nvShift) * C + (size_t)src * C + c];
        }
        Alds[r * K4 + k] = v;
    }
    __syncthreads();

    const int lane    = threadIdx.x & 31;
    const int wave    = threadIdx.x >> 5;
    const int nwaves  = blockDim.x >> 5;
    const int halfSel = lane >> 4;             // 0: lanes 0-15, 1: lanes 16-31
    const int lm      = lane & 15;
    const int nTiles  = Cout >> 4;

    for (int nt = wave; nt < nTiles; nt += nwaves) {   // wave-uniform loop
        const int col = nt * 16 + lm;
        const float* __restrict__ Wp = W + col;
        const float bv = bias[col];
        v8f acc0, acc1;
        #pragma unroll
        for (int v = 0; v < 8; ++v) { acc0[v] = bv; acc1[v] = 0.0f; }

        const float* aRow = Alds + lm * K4 + halfSel * 2;
        #pragma unroll 4
        for (int k0 = 0; k0 < K4; k0 += 8) {
            const int ks0 = k0 + halfSel * 2;
            // A 16x4 f32 frag: lane L -> M = L%16 ; vgpr v -> K = 2*(L/16)+v
            v2f a0 = *(const v2f*)(aRow + k0);
            // B 4x16 f32 frag: lane L -> N = L%16 ; vgpr v -> K = 2*(L/16)+v
            v2f b0;
            b0.x = (ks0 + 0 < K) ? Wp[(size_t)(ks0 + 0) * Cout] : 0.0f;
            b0.y = (ks0 + 1 < K) ? Wp[(size_t)(ks0 + 1) * Cout] : 0.0f;
            acc0 = __builtin_amdgcn_wmma_f32_16x16x4_f32(
                false, a0, false, b0, (short)0, acc0, false, false);

            if (k0 + 4 < K4) {
                const int ks1 = ks0 + 4;
                v2f a1 = *(const v2f*)(aRow + k0 + 4);
                v2f b1;
                b1.x = (ks1 + 0 < K) ? Wp[(size_t)(ks1 + 0) * Cout] : 0.0f;
                b1.y = (ks1 + 1 < K) ? Wp[(size_t)(ks1 + 1) * Cout] : 0.0f;
                acc1 = __builtin_amdgcn_wmma_f32_16x16x4_f32(
                    false, a1, false, b1, (short)0, acc1, false, false);
            }
        }

        // ELU + store; C/D layout: vgpr v, halfSel -> M = v + 8*halfSel ; N = lm
        #pragma unroll
        for (int v = 0; v < 8; ++v) {
            const float d = acc0[v] + acc1[v];
            const float o = d > 0.0f ? d : expm1f(d);
            const int r = v + halfSel * 8;
            y[(((size_t)b << nvShift) + (n0 + r)) * Cout + col] = o;
        }
    }
}

// ---------------------------------------------------------------------------
// out[b, row[k], c] += val[k] * xin[b, col[k], c]
// ---------------------------------------------------------------------------
__global__ void pool_scatter(const float* __restrict__ xin,  // [B, Nin, C]
                             const int* __restrict__ rowI,
                             const int* __restrict__ colI,
                             const float* __restrict__ val,
                             float* __restrict__ xout,       // [B, Nout, C] (zeroed)
                             int nnz, int Nin, int Nout, int C)
{
    const size_t tid = (size_t)blockIdx.x * blockDim.x + threadIdx.x;
    const size_t total = (size_t)16 * nnz * C;
    if (tid >= total) return;
    const int c = (int)(tid % C);
    const size_t t = tid / C;
    const int k = (int)(t % nnz);
    const int b = (int)(t / nnz);
    const float w = val[k] * xin[((size_t)b * Nin + colI[k]) * C + c];
    atomicAdd(&xout[((size_t)b * Nout + rowI[k]) * C + c], w);
}

__global__ void zero_f32(float* __restrict__ p, size_t n)
{
    const size_t i = (size_t)blockIdx.x * blockDim.x + threadIdx.x;
    if (i < n) p[i] = 0.0f;
}

__global__ void init_out_bias(float* __restrict__ out, const float* __restrict__ bf)
{
    const int i = blockIdx.x * blockDim.x + threadIdx.x;
    if (i < 16 * 256) out[i] = bf[i & 255];
}

// ---------------------------------------------------------------------------
// Final: out[16,256] += A[16,65536] @ Wf[65536,256], K split across blocks.
// Block stages A[16, FG_KCHUNK] in LDS; 16 waves own one 16-col tile each.
// ---------------------------------------------------------------------------
__global__ void final_gemm_wmma(const float* __restrict__ A,   // [16, Ktot]
                                const float* __restrict__ Wf,  // [Ktot, 256]
                                float* __restrict__ out,       // [16, 256]
                                int Ktot)
{
    __shared__ float Alds[16 * FG_KCHUNK];
    const int kStart = blockIdx.x * FG_KCHUNK;

    for (int e = threadIdx.x; e < 16 * FG_KCHUNK; e += blockDim.x) {
        const int r = e / FG_KCHUNK, k = e % FG_KCHUNK;
        Alds[e] = A[(size_t)r * Ktot + kStart + k];
    }
    __syncthreads();

    const int lane    = threadIdx.x & 31;
    const int wave    = threadIdx.x >> 5;      // 0..15 -> column tile
    const int halfSel = lane >> 4;
    const int lm      = lane & 15;
    const int col     = wave * 16 + lm;
    const float* __restrict__ Wp = Wf + col;

    v8f acc0, acc1;
    #pragma unroll
    for (int v = 0; v < 8; ++v) { acc0[v] = 0.0f; acc1[v] = 0.0f; }

    const float* aRow = Alds + lm * FG_KCHUNK + halfSel * 2;
    #pragma unroll 4
    for (int kk = 0; kk < FG_KCHUNK; kk += 8) {
        const size_t ks0 = (size_t)(kStart + kk + halfSel * 2);
        v2f a0 = *(const v2f*)(aRow + kk);
        v2f b0;
        b0.x = Wp[(ks0 + 0) * 256];
        b0.y = Wp[(ks0 + 1) * 256];
        acc0 = __builtin_amdgcn_wmma_f32_16x16x4_f32(
            false, a0, false, b0, (short)0, acc0, false, false);

        v2f a1 = *(const v2f*)(aRow + kk + 4);
        v2f b1;
        b1.x = Wp[(ks0 + 4) * 256];
        b1.y = Wp[(ks0 + 5) * 256];
        acc1 = __builtin_amdgcn_wmma_f32_16x16x4_f32(
            false, a1, false, b1, (short)0, acc1, false, false);
    }

    #pragma unroll
    for (int v = 0; v < 8; ++v) {
        const int brow = v + halfSel * 8;      // M == batch index
        atomicAdd(&out[brow * 256 + col], acc0[v] + acc1[v]);
    }
}

// ---------------------------------------------------------------------------
extern "C" void kernel_launch(void* const* d_in, const int* in_sizes, int n_in,
                              void* d_out, int out_size, void* d_ws, size_t ws_size,
                              hipStream_t stream)
{
    (void)in_sizes; (void)n_in; (void)out_size; (void)ws_size;
    static const int VERTS[5]   = {65536, 16384, 4096, 1024, 256};
    static const int VSHIFT[5]  = {16, 14, 12, 10, 8};
    static const int CH[5]      = {3, 32, 64, 128, 256};
    const int S = 9, B = 16;

    const float* x = (const float*)d_in[0];
    const int   *idx[4], *rowI[4], *colI[4];
    const float *val[4], *W[4], *bias[4];
    for (int i = 0; i < 4; ++i) {
        idx[i]  = (const int*)  d_in[1 + 6 * i];
        rowI[i] = (const int*)  d_in[2 + 6 * i];
        colI[i] = (const int*)  d_in[3 + 6 * i];
        val[i]  = (const float*)d_in[4 + 6 * i];
        W[i]    = (const float*)d_in[5 + 6 * i];
        bias[i] = (const float*)d_in[6 + 6 * i];
    }
    const float* Wf  = (const float*)d_in[25];
    const float* bfv = (const float*)d_in[26];
    float* out = (float*)d_out;

    float* bufA = (float*)d_ws;                              // <= 16*65536*32 f32
    float* bufB = bufA + (size_t)B * 65536 * 32;             // <= 16*16384*32 f32

    const float* cur = x;
    for (int i = 0; i < 4; ++i) {
        const int Nv = VERTS[i], C = CH[i], Co = CH[i + 1];
        const int K = S * C, K4 = (K + 3) & ~3;
        const int rowTiles = (B * Nv) / 16;                  // == Nv
        const int nTiles = Co / 16;
        const int waves = nTiles < 4 ? nTiles : 4;
        const size_t ldsBytes = (size_t)16 * K4 * sizeof(float);
        spiral_conv_wmma<<<rowTiles, 32 * waves, ldsBytes, stream>>>(
            cur, idx[i], W[i], bias[i], bufA, VSHIFT[i], C, S, Co, K4);

        const int Nout = VERTS[i + 1];
        const size_t outElems = (size_t)B * Nout * Co;
        zero_f32<<<(int)((outElems + 255) / 256), 256, 0, stream>>>(bufB, outElems);

        const int nnz = 3 * Nout;
        const size_t total = (size_t)B * nnz * Co;
        pool_scatter<<<(int)((total + 255) / 256), 256, 0, stream>>>(
            bufA, rowI[i], colI[i], val[i], bufB, nnz, Nv, Nout, Co);
        cur = bufB;
    }

    init_out_bias<<<(16 * 256 + 255) / 256, 256, 0, stream>>>(out, bfv);
    final_gemm_wmma<<<65536 / FG_KCHUNK, 512, 0, stream>>>(cur, Wf, out, 65536);
}